// SparseGroup_20461224198847
// MI455X (gfx1250) — compile-verified
//
#include <hip/hip_runtime.h>

// Problem constants (match reference)
#define BATCH 4
#define NPTS  32768
#define G     512
#define KNN   32
#define FD    128

// d_out layout (float offsets), tuple concatenated in return order:
// neighborhoods [B,G,K,3], centers [B,G,3], batch_idxs [B,G,K],
// pts [B,N,3], input_features [B,G,K,FD]
#define OUT_NEIGH 0
#define OUT_CENT  (BATCH * G * KNN * 3)          // 196608
#define OUT_BIDX  (OUT_CENT + BATCH * G * 3)     // 202752
#define OUT_PTS   (OUT_BIDX + BATCH * G * KNN)   // 268288
#define OUT_FEAT  (OUT_PTS + BATCH * NPTS * 3)   // 661504

typedef __attribute__((ext_vector_type(2))) float        v2f;
typedef __attribute__((ext_vector_type(8))) float        v8f;
typedef __attribute__((ext_vector_type(4))) unsigned int v4u;
typedef __attribute__((ext_vector_type(8))) int          v8i_t;
typedef __attribute__((ext_vector_type(4))) int          v4i_t;

// ---------------------------------------------------------------------------
// Kernel 1: strip batch column, emit pts output, init FPS mindist.
// Pure streaming: bandwidth bound, coalesced.
// ---------------------------------------------------------------------------
__global__ void __launch_bounds__(256)
prep_kernel(const float* __restrict__ xyz, float* __restrict__ pts_out,
            float* __restrict__ mindist)
{
    const int i = blockIdx.x * 256 + threadIdx.x;   // < BATCH*NPTS
    const float x = xyz[i * 4 + 1];
    const float y = xyz[i * 4 + 2];
    const float z = xyz[i * 4 + 3];
    pts_out[i * 3 + 0] = x;
    pts_out[i * 3 + 1] = y;
    pts_out[i * 3 + 2] = z;
    mindist[i] = 1e10f;
}

// ---------------------------------------------------------------------------
// Kernel 2: furthest point sampling. One block per cloud, 256 threads.
// 511 sequential rounds: distance-to-last-center pass + argmax LDS reduction.
// First-max tie-break (lowest index) to match jnp.argmax.
// ---------------------------------------------------------------------------
__global__ void __launch_bounds__(256)
fps_kernel(const float* __restrict__ pts, float* __restrict__ mindist,
           float* __restrict__ cent_out, float* __restrict__ centers4,
           float* __restrict__ cn2)
{
    __shared__ float sb[256];
    __shared__ int   si[256];
    __shared__ float scx, scy, scz;

    const int b = blockIdx.x;
    const int t = threadIdx.x;
    const float* p  = pts     + (size_t)b * NPTS * 3;
    float*       md = mindist + (size_t)b * NPTS;

    if (t == 0) {
        const float x = p[0], y = p[1], z = p[2];
        scx = x; scy = y; scz = z;
        float* co = cent_out + (size_t)b * G * 3;
        co[0] = x; co[1] = y; co[2] = z;
        float* c4 = centers4 + (size_t)b * G * 4;
        c4[0] = x; c4[1] = y; c4[2] = z; c4[3] = 0.0f;
        cn2[b * G] = x * x + y * y + z * z;
    }
    __syncthreads();

    for (int g = 1; g < G; ++g) {
        const float cx = scx, cy = scy, cz = scz;
        float best = -1.0f;
        int   bidx = 0;
        for (int j = t; j < NPTS; j += 256) {
            const float dx = p[j * 3 + 0] - cx;
            const float dy = p[j * 3 + 1] - cy;
            const float dz = p[j * 3 + 2] - cz;
            const float d  = dx * dx + dy * dy + dz * dz;
            float m = md[j];
            m = fminf(m, d);
            md[j] = m;
            if (m > best) { best = m; bidx = j; }   // strict > keeps first max
        }
        sb[t] = best; si[t] = bidx;
        __syncthreads();
        for (int s = 128; s > 0; s >>= 1) {
            if (t < s) {
                const float ob = sb[t + s];
                const int   oi = si[t + s];
                if (ob > sb[t] || (ob == sb[t] && oi < si[t])) {
                    sb[t] = ob; si[t] = oi;
                }
            }
            __syncthreads();
        }
        if (t == 0) {
            const int idx = si[0];
            const float x = p[idx * 3 + 0];
            const float y = p[idx * 3 + 1];
            const float z = p[idx * 3 + 2];
            scx = x; scy = y; scz = z;
            float* co = cent_out + ((size_t)b * G + g) * 3;
            co[0] = x; co[1] = y; co[2] = z;
            float* c4 = centers4 + ((size_t)b * G + g) * 4;
            c4[0] = x; c4[1] = y; c4[2] = z; c4[3] = 0.0f;
            cn2[b * G + g] = x * x + y * y + z * z;
        }
        __syncthreads();
    }
}

// ---------------------------------------------------------------------------
// Tensor Data Mover: DMA one contiguous strip of CHUNK*3 floats (6 KB) of
// point coordinates into LDS. D# group0: count=1, lds_addr, global tile
// address, type=2. Group1: data_size=4B, 1-row tile of CHUNK*3 elements.
// Tracked with TENSORcnt (in-order per wave).
// ---------------------------------------------------------------------------
#define CHUNK  512                  // points per LDS stage
#define TILES  (CHUNK / 16)         // 32 WMMA tiles per chunk
#define NCHUNK (NPTS / CHUNK)       // 64

__device__ __forceinline__ void tdm_load_pts_chunk(const float* gsrc,
                                                   unsigned lds_byte_off)
{
    const unsigned long long ga = (unsigned long long)gsrc;
    v4u g0;
    g0[0] = 1u;                                       // count = 1 (valid D#)
    g0[1] = lds_byte_off;                             // lds_addr (bytes)
    g0[2] = (unsigned)ga;                             // global_addr[31:0]
    g0[3] = (unsigned)((ga >> 32) & 0x01ffffffu)      // global_addr[56:32]
          | (2u << 30);                               // type = 2 ("image")
    v8i_t g1;
    g1[0] = (int)(2u << 16);                          // data_size = 4 bytes
    g1[1] = (int)0x80000000u;                         // tensor_dim0[15:0]=0x8000 (98304 lo)
    g1[2] = 0x00010001;                               // tensor_dim0[31:16]=1, tensor_dim1=1
    g1[3] = (int)((CHUNK * 3) << 16);                 // tile_dim0 = 1536
    g1[4] = 1;                                        // tile_dim1 = 1
    g1[5] = NPTS * 3;                                 // tensor_dim0_stride = 98304
    g1[6] = 0;
    g1[7] = 0;
    const v4i_t z4 = {0, 0, 0, 0};                    // groups 2/3 unused (<=2D)
    const v8i_t z8 = {0, 0, 0, 0, 0, 0, 0, 0};
    __builtin_amdgcn_tensor_load_to_lds(g0, g1, z4, z4, z8, 0);
}

// ---------------------------------------------------------------------------
// Kernel 3: KNN via V_WMMA_F32_16X16X4_F32 with TDM-double-buffered point
// streaming. One wave per block, 16 centers per block. A = centers (16x4,
// zero-padded K). B = 16-point tile read from the LDS stage (branchless
// cndmask selects -> no EXEC juggling). d2 = fma(-2, dot, |c|^2 + |p|^2);
// distances staged in LDS (stride 17, conflict-free); 16 lanes keep a
// per-center sorted top-32 list.
// ---------------------------------------------------------------------------
#define SDIST_FLOATS (CHUNK * 17)            // 8704 floats = 34816 B
#define SPTS_OFF     SDIST_FLOATS            // float index of pts staging area
#define SPTS_FLOATS  (CHUNK * 3)             // 1536 floats = 6144 B per buffer

__global__ void __launch_bounds__(32)
knn_kernel(const float* __restrict__ pts, const float* __restrict__ centers4,
           const float* __restrict__ cn2, int* __restrict__ knn_out)
{
    __shared__ float smem[SDIST_FLOATS + 2 * SPTS_FLOATS];   // 47104 B
    float* sdist = smem;

    const int lane = threadIdx.x;
    const int half = lane >> 4;           // 0: lanes 0-15, 1: lanes 16-31
    const int lm   = lane & 15;
    const int blocksPerCloud = G / 16;    // 32
    const int b     = blockIdx.x / blocksPerCloud;
    const int cg    = blockIdx.x % blocksPerCloud;
    const int cbase = cg * 16;

    // A-matrix (16x4 f32): lanes 0-15 hold (K=0,K=1)=(x,y) of row lm,
    // lanes 16-31 hold (K=2,K=3)=(z,0) of row lm (per ISA 32-bit A layout).
    const float* c4 = centers4 + (size_t)(b * G + cbase + lm) * 4;
    v2f A;
    A[0] = half ? c4[2] : c4[0];
    A[1] = half ? 0.0f  : c4[1];

    // |c|^2 for the 8 center rows this lane's C-VGPRs cover.
    float cn2v[8];
#pragma unroll
    for (int r = 0; r < 8; ++r)
        cn2v[r] = cn2[b * G + cbase + half * 8 + r];

    float bd[KNN];
    int   bi[KNN];
#pragma unroll
    for (int k = 0; k < KNN; ++k) { bd[k] = 3.4e38f; bi[k] = 0; }
    float worst = 3.4e38f;

    const float* ppts = pts + (size_t)b * NPTS * 3;

    // Prime the pipeline: DMA chunk 0 into buffer 0.
    tdm_load_pts_chunk(ppts, (unsigned)(SPTS_OFF * 4));

    for (int chunk = 0; chunk < NCHUNK; ++chunk) {
        const int p0c = chunk * CHUNK;
        const int cur = chunk & 1;

        if (chunk + 2 < NCHUNK)   // warm GL2 for the DMA two chunks out
            __builtin_prefetch(ppts + (size_t)(p0c + 2 * CHUNK) * 3 + lane * 48,
                               0, 1);

        if (chunk + 1 < NCHUNK) {
            // Kick DMA for the next chunk into the other buffer, then wait
            // until only that newest transfer is outstanding (loads complete
            // in order) -> current chunk's data is in LDS.
            tdm_load_pts_chunk(ppts + (size_t)(p0c + CHUNK) * 3,
                               (unsigned)((SPTS_OFF + (cur ^ 1) * SPTS_FLOATS) * 4));
            __builtin_amdgcn_s_wait_tensorcnt(1);
        } else {
            __builtin_amdgcn_s_wait_tensorcnt(0);
        }
        asm volatile("" ::: "memory");   // keep LDS reads below the wait

        const float* sp = smem + SPTS_OFF + cur * SPTS_FLOATS;

#pragma unroll 4
        for (int tile = 0; tile < TILES; ++tile) {
            const int t0 = tile * 16;
            // All lanes read point (t0+lm): x,y,z from the LDS stage.
            const float x = sp[(t0 + lm) * 3 + 0];
            const float y = sp[(t0 + lm) * 3 + 1];
            const float z = sp[(t0 + lm) * 3 + 2];
            // B-matrix (4x16 f32): lanes 0-15 hold (K0,K1)=(x,y), lanes 16-31
            // hold (K2,K3)=(z,0) -- branchless selects, EXEC stays full.
            v2f Bv;
            Bv[0] = half ? z    : x;
            Bv[1] = half ? 0.0f : y;

            v8f C = {};
            C = __builtin_amdgcn_wmma_f32_16x16x4_f32(
                    false, A, false, Bv, (short)0, C, false, false);

            const float pn = x * x + y * y + z * z;   // |p|^2 of this column
#pragma unroll
            for (int r = 0; r < 8; ++r) {
                const int   m  = half * 8 + r;
                const float d2 = __builtin_fmaf(-2.0f, C[r], cn2v[r] + pn);
                sdist[(t0 + lm) * 17 + m] = d2;
            }
        }
        __syncthreads();

        // Selection: lane c scans this chunk for center c (consecutive lanes ->
        // consecutive LDS banks).
        if (lane < 16) {
            for (int t = 0; t < CHUNK; ++t) {
                const float d = sdist[t * 17 + lane];
                if (d < worst) {
                    int j = KNN - 1;
                    while (j > 0 && bd[j - 1] > d) {
                        bd[j] = bd[j - 1]; bi[j] = bi[j - 1]; --j;
                    }
                    bd[j] = d; bi[j] = p0c + t;
                    worst = bd[KNN - 1];
                }
            }
        }
        __syncthreads();
    }

    if (lane < 16) {
        int* o = knn_out + (size_t)(b * G + cbase + lane) * KNN;
#pragma unroll
        for (int k = 0; k < KNN; ++k) o[k] = bi[k];
    }
}

// ---------------------------------------------------------------------------
// Kernel 4: gathers. One block per (b,g,k) row; 128 threads copy the feature
// row coalesced; threads 0-2 emit the re-centered neighborhood; thread 0 the
// index (as float, exactly representable).
// ---------------------------------------------------------------------------
__global__ void __launch_bounds__(128)
gather_kernel(const float* __restrict__ feature, const float* __restrict__ pts,
              const float* __restrict__ cent, const int* __restrict__ knn,
              float* __restrict__ neigh, float* __restrict__ bidx,
              float* __restrict__ feat)
{
    const int r   = blockIdx.x;                 // [0, BATCH*G*KNN)
    const int t   = threadIdx.x;
    const int b   = r / (G * KNN);
    const int rem = r - b * (G * KNN);
    const int g   = rem / KNN;
    const int idx = knn[r];

    feat[(size_t)r * FD + t] = feature[((size_t)b * NPTS + idx) * FD + t];
    if (t < 3)
        neigh[(size_t)r * 3 + t] =
            pts[((size_t)b * NPTS + idx) * 3 + t] -
            cent[((size_t)b * G + g) * 3 + t];
    if (t == 0)
        bidx[r] = (float)idx;
}

// ---------------------------------------------------------------------------
extern "C" void kernel_launch(void* const* d_in, const int* in_sizes, int n_in,
                              void* d_out, int out_size, void* d_ws, size_t ws_size,
                              hipStream_t stream)
{
    const float* xyz     = (const float*)d_in[0];
    const float* feature = (const float*)d_in[1];

    float* out   = (float*)d_out;
    float* neigh = out + OUT_NEIGH;
    float* cent  = out + OUT_CENT;
    float* bidx  = out + OUT_BIDX;
    float* ptsO  = out + OUT_PTS;
    float* featO = out + OUT_FEAT;

    // Workspace carve-up (~0.85 MB)
    float* ws       = (float*)d_ws;
    float* mindist  = ws;                              // B*N
    float* centers4 = ws + BATCH * NPTS;               // B*G*4 (xyz0)
    float* cn2      = centers4 + BATCH * G * 4;        // B*G
    int*   knnI     = (int*)(cn2 + BATCH * G);         // B*G*KNN ints

    prep_kernel  <<<BATCH * NPTS / 256, 256, 0, stream>>>(xyz, ptsO, mindist);
    fps_kernel   <<<BATCH,              256, 0, stream>>>(ptsO, mindist, cent, centers4, cn2);
    knn_kernel   <<<BATCH * (G / 16),    32, 0, stream>>>(ptsO, centers4, cn2, knnI);
    gather_kernel<<<BATCH * G * KNN,    128, 0, stream>>>(feature, ptsO, cent, knnI,
                                                          neigh, bidx, featO);
}